// GCNConv_25829933318158
// MI455X (gfx1250) — compile-verified
//
#include <hip/hip_runtime.h>

typedef __attribute__((ext_vector_type(2))) float v2f;
typedef __attribute__((ext_vector_type(8))) float v8f;

#define IN_F  256
#define OUT_F 128

// ---------------------------------------------------------------------------
// Phase 2 (listed first so disasm shows the atomic path):
// out[dst] += val * h[src]   (h and out both live in the 192MB L2)
// One wave per edge: 32 lanes x float4 = 128 features.
// Gather: global_load_b128 ; scatter: hardware global_atomic_add_f32.
// ---------------------------------------------------------------------------
__global__ __launch_bounds__(256) void gcn_spmm_scatter(
    const float* __restrict__ h, const int* __restrict__ esrc,
    const int* __restrict__ edst, const float* __restrict__ evals,
    float* __restrict__ out, int n_edges)
{
    __shared__ int   s_src[8];
    __shared__ int   s_dst[8];
    __shared__ float s_val[8];

    const int tid   = threadIdx.x;
    const int ebase = blockIdx.x * 8;

    if (tid < 8) {
        int e   = ebase + tid;
        bool ok = e < n_edges;
        s_src[tid] = ok ? esrc[e]  : 0;
        s_dst[tid] = ok ? edst[e]  : -1;
        s_val[tid] = ok ? evals[e] : 0.f;
    }
    __syncthreads();

    const int wave = tid >> 5;
    const int lane = tid & 31;

    const int d = s_dst[wave];
    if (d < 0) return;
    const int   s = s_src[wave];
    const float v = s_val[wave];

    float4 m = *((const float4*)(h + (size_t)s * OUT_F) + lane);

    float* op = out + (size_t)d * OUT_F + lane * 4;
    unsafeAtomicAdd(op + 0, v * m.x);
    unsafeAtomicAdd(op + 1, v * m.y);
    unsafeAtomicAdd(op + 2, v * m.z);
    unsafeAtomicAdd(op + 3, v * m.w);
}

// ---------------------------------------------------------------------------
// Phase 1: h[N,128] = x[N,256] @ W[256,128] + b  via V_WMMA_F32_16X16X4_F32
// One wave -> one 16x16 tile. Block = 8 waves = 16-row strip x 128 cols.
// A strip (16x256 fp32 = 16KB) staged in LDS once per block, padded stride.
// ---------------------------------------------------------------------------
__global__ __launch_bounds__(256) void gcn_gemm_wmma(
    const float* __restrict__ x, const float* __restrict__ w,
    const float* __restrict__ bias, float* __restrict__ h, int n_nodes)
{
    __shared__ float As[16][260];   // 260: 256 % 64banks == 0 -> pad by 4

    const int tid  = threadIdx.x;
    const int row0 = blockIdx.x * 16;
    const bool full_strip = (row0 + 16) <= n_nodes;   // uniform across block

    // Cooperative load of the 16x256 A strip as float4s (256 thr * 4 f4 each)
    for (int i = tid; i < 16 * 64; i += 256) {
        int r  = i >> 6;          // row within strip
        int cc = (i & 63) * 4;    // column
        int row = row0 + r;
        float4 val = make_float4(0.f, 0.f, 0.f, 0.f);
        if (full_strip || row < n_nodes)
            val = *(const float4*)(x + (size_t)row * IN_F + cc);
        *(float4*)&As[r][cc] = val;   // row stride 1040B: 16B aligned
    }
    __syncthreads();

    const int wave = tid >> 5;          // 0..7 -> 16-col tile
    const int lane = tid & 31;
    const int half = lane >> 4;         // 0: lanes 0-15, 1: lanes 16-31
    const int l16  = lane & 15;
    const int col0 = wave * 16;

    v8f c = {};

    // K loop: 256 / 4 = 64 WMMA steps, 8 per iteration
    #pragma unroll 8
    for (int k = 0; k < IN_F; k += 4) {
        const int k0 = k + half * 2;
        // A 16x4 fragment: VGPR0 = A[m][k0], VGPR1 = A[m][k0+1], m = lane%16
        v2f a;
        a.x = As[l16][k0 + 0];
        a.y = As[l16][k0 + 1];
        // B 4x16 fragment: VGPR0 = W[k0][n], VGPR1 = W[k0+1][n], n = col0+lane%16
        v2f b;
        b.x = w[(size_t)(k0 + 0) * OUT_F + col0 + l16];
        b.y = w[(size_t)(k0 + 1) * OUT_F + col0 + l16];
        c = __builtin_amdgcn_wmma_f32_16x16x4_f32(
                /*neg_a=*/false, a, /*neg_b=*/false, b,
                /*c_mod=*/(short)0, c, /*reuse_a=*/false, /*reuse_b=*/false);
    }

    // C/D layout: VGPR r -> row = row0 + r + 8*half, col = col0 + lane%16
    const float bcol = bias[col0 + l16];
    float* hp = h + (size_t)(row0 + 8 * half) * OUT_F + col0 + l16;
    if (full_strip) {
        #pragma unroll
        for (int r = 0; r < 8; ++r)
            hp[(size_t)r * OUT_F] = c[r] + bcol;
    } else {
        #pragma unroll
        for (int r = 0; r < 8; ++r) {
            int row = row0 + r + 8 * half;
            if (row < n_nodes)
                h[(size_t)row * OUT_F + col0 + l16] = c[r] + bcol;
        }
    }
}

// ---------------------------------------------------------------------------
extern "C" void kernel_launch(void* const* d_in, const int* in_sizes, int n_in,
                              void* d_out, int out_size, void* d_ws, size_t ws_size,
                              hipStream_t stream)
{
    const float* x         = (const float*)d_in[0];
    const int*   edge_src  = (const int*)  d_in[1];
    const int*   edge_dst  = (const int*)  d_in[2];
    const float* edge_vals = (const float*)d_in[3];
    const float* weight    = (const float*)d_in[4];
    const float* bias      = (const float*)d_in[5];
    float*       out       = (float*)d_out;
    float*       h         = (float*)d_ws;   // N_NODES * OUT_F fp32 (51.2 MB)

    const int n_nodes = in_sizes[0] / IN_F;
    const int n_edges = in_sizes[1];

    // out must start at zero for the scatter-add (bias lives inside h)
    hipMemsetAsync(d_out, 0, (size_t)out_size * sizeof(float), stream);

    gcn_gemm_wmma<<<(n_nodes + 15) / 16, 256, 0, stream>>>(x, weight, bias, h, n_nodes);

    const int blocks = (n_edges + 7) / 8;
    gcn_spmm_scatter<<<blocks, 256, 0, stream>>>(h, edge_src, edge_dst, edge_vals,
                                                 out, n_edges);
}